// GNNClassifier_83305185673529
// MI455X (gfx1250) — compile-verified
//
#include <hip/hip_runtime.h>
#include <hip/hip_bf16.h>

#define N_NODES 50000
#define N_EDGES 800000
#define IN_CH   128
#define HID     256
#define OUT_CH  10
#define N_GRAPHS 64

typedef float v2f __attribute__((ext_vector_type(2)));
typedef float v8f __attribute__((ext_vector_type(8)));

// ---------------------------------------------------------------------------
// Utility: zero a float buffer (ws is poisoned with 0xAA by the harness).
// ---------------------------------------------------------------------------
__global__ void zero_f32(float* __restrict__ p, int n) {
  int i = blockIdx.x * blockDim.x + threadIdx.x;
  if (i < n) p[i] = 0.0f;
}

// ---------------------------------------------------------------------------
// Degree: deg[dst] += 1 per edge (fp32, matches reference segment_sum of ones)
// ---------------------------------------------------------------------------
__global__ void degree_kernel(const int* __restrict__ dst, float* __restrict__ deg) {
  int e = blockIdx.x * blockDim.x + threadIdx.x;
  if (e < N_EDGES) unsafeAtomicAdd(&deg[dst[e]], 1.0f);
}

// ---------------------------------------------------------------------------
// Edge scatter: agg[dst, :] += h[src, :]   (float4 chunks, L2-resident atomics)
// agg has row stride HID (=256) always; h has row stride ld_in.
// ---------------------------------------------------------------------------
__global__ void scatter_kernel(const float* __restrict__ h, int ld_in, int nchunk4,
                               const int* __restrict__ src, const int* __restrict__ dst,
                               float* __restrict__ agg) {
  long long i = (long long)blockIdx.x * blockDim.x + threadIdx.x;
  long long total = (long long)N_EDGES * nchunk4;
  if (i >= total) return;
  int e  = (int)(i / nchunk4);
  int c4 = (int)(i % nchunk4);
  int s = src[e];
  int d = dst[e];
  const float4 v = *(const float4*)(h + (long long)s * ld_in + c4 * 4);
  float* o = agg + (long long)d * HID + c4 * 4;
  unsafeAtomicAdd(o + 0, v.x);
  unsafeAtomicAdd(o + 1, v.y);
  unsafeAtomicAdd(o + 2, v.z);
  unsafeAtomicAdd(o + 3, v.w);
}

// ---------------------------------------------------------------------------
// Fused SAGEConv tile kernel (fp32 WMMA 16x16x4):
//   out[i,:] = relu( (agg[i,:]/max(deg,1)) @ Wl.T + bl + h_in[i,:] @ Wr.T )
// written IN PLACE into the agg buffer (each block exclusively owns 16 rows).
// Block = 128 threads = 4 waves; block tile = 16 nodes x 256 out channels;
// each wave computes 4 N-tiles of 16 via v_wmma_f32_16x16x4_f32.
// ---------------------------------------------------------------------------
#define SASTRIDE 260  // 256 + 4 pad -> conflict-free column reads (4m mod 64 distinct)

__global__ __launch_bounds__(128)
void fused_sage(float* __restrict__ agg_out,        // [N, 256] : in=agg, out=h_next
                const float* __restrict__ h_in,     // [N, ld_in]
                const float* __restrict__ deg,      // [N]
                const float* __restrict__ Wl,       // [256, Kdim] row-major
                const float* __restrict__ bl,       // [256]
                const float* __restrict__ Wr,       // [256, Kdim] row-major
                int Kdim, int ld_in) {
  __shared__ float sA[16 * SASTRIDE];
  __shared__ float sX[16 * SASTRIDE];

  const int tid   = threadIdx.x;
  const int mbase = blockIdx.x * 16;           // 50000 = 3125 * 16 exactly

  // Cooperative staged load: 16 rows x Kdim floats each (float4), scale agg by 1/deg.
  const int rowc4 = Kdim >> 2;
  const int nchunks = 16 * rowc4;
  for (int i = tid; i < nchunks; i += 128) {
    int r  = i / rowc4;
    int c4 = i % rowc4;
    int row = mbase + r;
    float invd = 1.0f / fmaxf(deg[row], 1.0f);
    float4 a = *(const float4*)(agg_out + (long long)row * HID + c4 * 4);
    float4 x = *(const float4*)(h_in   + (long long)row * ld_in + c4 * 4);
    float* pa = sA + r * SASTRIDE + c4 * 4;
    pa[0] = a.x * invd; pa[1] = a.y * invd; pa[2] = a.z * invd; pa[3] = a.w * invd;
    float* px = sX + r * SASTRIDE + c4 * 4;
    px[0] = x.x; px[1] = x.y; px[2] = x.z; px[3] = x.w;
  }
  __syncthreads();

  const int lane = tid & 31;
  const int wid  = tid >> 5;
  const int half = lane >> 4;   // 0: K={0,1}; 1: K={2,3}  (ISA 32-bit A 16x4 layout)
  const int l15  = lane & 15;

  for (int t = 0; t < 4; ++t) {
    const int nbase = (wid * 4 + t) * 16;
    // C/D layout: VGPR r <-> row (r + 8*half), col l15. Bias same for all rows.
    float bias = bl[nbase + l15];
    v8f acc;
#pragma unroll
    for (int r = 0; r < 8; ++r) acc[r] = bias;

    const float* wlrow = Wl + (long long)(nbase + l15) * Kdim;  // B[k][n] = Wl[n][k]
    const float* wrrow = Wr + (long long)(nbase + l15) * Kdim;
    const float* arow  = sA + l15 * SASTRIDE;
    const float* xrow  = sX + l15 * SASTRIDE;

#pragma unroll 4
    for (int k = 0; k < Kdim; k += 4) {
      const int ko = k + 2 * half;
      v2f a, b;
      // --- agg @ Wl.T ---
      a.x = arow[ko]; a.y = arow[ko + 1];
      {
        const float2 w = *(const float2*)(wlrow + ko);
        b.x = w.x; b.y = w.y;
      }
      acc = __builtin_amdgcn_wmma_f32_16x16x4_f32(false, a, false, b,
                                                  (short)0, acc, false, false);
      // --- h_in @ Wr.T (accumulate into same C) ---
      a.x = xrow[ko]; a.y = xrow[ko + 1];
      {
        const float2 w = *(const float2*)(wrrow + ko);
        b.x = w.x; b.y = w.y;
      }
      acc = __builtin_amdgcn_wmma_f32_16x16x4_f32(false, a, false, b,
                                                  (short)0, acc, false, false);
    }

    // ReLU + in-place store (block owns rows mbase..mbase+15 exclusively)
#pragma unroll
    for (int r = 0; r < 8; ++r) {
      int row = mbase + r + 8 * half;
      agg_out[(long long)row * HID + nbase + l15] = fmaxf(acc[r], 0.0f);
    }
  }
}

// ---------------------------------------------------------------------------
// Global mean pool: pooled[batch[n], :] += h[n, :], cnt[batch[n]] += 1
// ---------------------------------------------------------------------------
__global__ void pool_scatter(const float* __restrict__ h, const int* __restrict__ batch,
                             float* __restrict__ pooled, float* __restrict__ cnt) {
  long long i = (long long)blockIdx.x * blockDim.x + threadIdx.x;
  long long total = (long long)N_NODES * (HID / 4);
  if (i >= total) return;
  int node = (int)(i >> 6);
  int c4   = (int)(i & 63);
  int b = batch[node];
  const float4 v = *(const float4*)(h + (long long)node * HID + c4 * 4);
  float* o = pooled + (long long)b * HID + c4 * 4;
  unsafeAtomicAdd(o + 0, v.x);
  unsafeAtomicAdd(o + 1, v.y);
  unsafeAtomicAdd(o + 2, v.z);
  unsafeAtomicAdd(o + 3, v.w);
  if (c4 == 0) unsafeAtomicAdd(&cnt[b], 1.0f);
}

// ---------------------------------------------------------------------------
// MLP head (tiny: 64 graphs): g1 = relu((pooled/cnt) @ lin1_W.T + b)
// ---------------------------------------------------------------------------
__global__ void head1(const float* __restrict__ pooled, const float* __restrict__ cnt,
                      const float* __restrict__ W, const float* __restrict__ bias,
                      float* __restrict__ g1) {
  int t = blockIdx.x * blockDim.x + threadIdx.x;   // 64 * 128
  if (t >= N_GRAPHS * 128) return;
  int g = t >> 7, j = t & 127;
  float invc = 1.0f / fmaxf(cnt[g], 1.0f);
  float acc = bias[j];
  const float* p = pooled + g * HID;
  const float* w = W + j * HID;
  for (int k = 0; k < HID; ++k) acc += (p[k] * invc) * w[k];
  g1[t] = fmaxf(acc, 0.0f);
}

__global__ void head2(const float* __restrict__ g1, const float* __restrict__ W,
                      const float* __restrict__ bias, float* __restrict__ out) {
  int t = blockIdx.x * blockDim.x + threadIdx.x;   // 64 * 10
  if (t >= N_GRAPHS * OUT_CH) return;
  int g = t / OUT_CH, o = t % OUT_CH;
  float acc = bias[o];
  const float* p = g1 + g * 128;
  const float* w = W + o * 128;
  for (int j = 0; j < 128; ++j) acc += p[j] * w[j];
  out[t] = acc;
}

// ---------------------------------------------------------------------------
// Launch
// ---------------------------------------------------------------------------
extern "C" void kernel_launch(void* const* d_in, const int* in_sizes, int n_in,
                              void* d_out, int out_size, void* d_ws, size_t ws_size,
                              hipStream_t stream) {
  const float* x     = (const float*)d_in[0];
  const int*   ei    = (const int*)d_in[1];   // [2, E] flat; harness maps integer -> int32
  const int*   batch = (const int*)d_in[2];
  const float* W1l = (const float*)d_in[3];
  const float* b1  = (const float*)d_in[4];
  const float* W1r = (const float*)d_in[5];
  const float* W2l = (const float*)d_in[6];
  const float* b2  = (const float*)d_in[7];
  const float* W2r = (const float*)d_in[8];
  const float* W3l = (const float*)d_in[9];
  const float* b3  = (const float*)d_in[10];
  const float* W3r = (const float*)d_in[11];
  const float* l1W = (const float*)d_in[12];
  const float* l1b = (const float*)d_in[13];
  const float* l2W = (const float*)d_in[14];
  const float* l2b = (const float*)d_in[15];
  float* out = (float*)d_out;

  // Workspace layout (floats): bufA, bufB (N x 256 each), deg, pooled, cnt, g1
  float* bufA   = (float*)d_ws;
  float* bufB   = bufA + (size_t)N_NODES * HID;
  float* deg    = bufB + (size_t)N_NODES * HID;
  float* pooled = deg + N_NODES;
  float* cnt    = pooled + (size_t)N_GRAPHS * HID;
  float* g1     = cnt + N_GRAPHS;

  const int* src = ei;
  const int* dst = ei + N_EDGES;

  const int ZB = 256;
  const int NH = N_NODES * HID;

  // degree (once)
  zero_f32<<<(N_NODES + ZB - 1) / ZB, ZB, 0, stream>>>(deg, N_NODES);
  degree_kernel<<<(N_EDGES + ZB - 1) / ZB, ZB, 0, stream>>>(dst, deg);

  // ---- Layer 1: agg(x)[128] -> bufA ; bufA := relu(bufA@W1l.T + b1 + x@W1r.T)
  zero_f32<<<(NH + ZB - 1) / ZB, ZB, 0, stream>>>(bufA, NH);
  {
    long long tot = (long long)N_EDGES * (IN_CH / 4);
    scatter_kernel<<<(int)((tot + ZB - 1) / ZB), ZB, 0, stream>>>(x, IN_CH, IN_CH / 4,
                                                                  src, dst, bufA);
  }
  fused_sage<<<N_NODES / 16, 128, 0, stream>>>(bufA, x, deg, W1l, b1, W1r, IN_CH, IN_CH);

  // ---- Layer 2: agg(h1=bufA) -> bufB ; bufB := relu(...)
  zero_f32<<<(NH + ZB - 1) / ZB, ZB, 0, stream>>>(bufB, NH);
  {
    long long tot = (long long)N_EDGES * (HID / 4);
    scatter_kernel<<<(int)((tot + ZB - 1) / ZB), ZB, 0, stream>>>(bufA, HID, HID / 4,
                                                                  src, dst, bufB);
  }
  fused_sage<<<N_NODES / 16, 128, 0, stream>>>(bufB, bufA, deg, W2l, b2, W2r, HID, HID);

  // ---- Layer 3: agg(h2=bufB) -> bufA ; bufA := relu(...)
  zero_f32<<<(NH + ZB - 1) / ZB, ZB, 0, stream>>>(bufA, NH);
  {
    long long tot = (long long)N_EDGES * (HID / 4);
    scatter_kernel<<<(int)((tot + ZB - 1) / ZB), ZB, 0, stream>>>(bufB, HID, HID / 4,
                                                                  src, dst, bufA);
  }
  fused_sage<<<N_NODES / 16, 128, 0, stream>>>(bufA, bufB, deg, W3l, b3, W3r, HID, HID);

  // ---- Pool + head
  zero_f32<<<(N_GRAPHS * HID + ZB - 1) / ZB, ZB, 0, stream>>>(pooled, N_GRAPHS * HID);
  zero_f32<<<1, ZB, 0, stream>>>(cnt, N_GRAPHS);
  {
    long long tot = (long long)N_NODES * (HID / 4);
    pool_scatter<<<(int)((tot + ZB - 1) / ZB), ZB, 0, stream>>>(bufA, batch, pooled, cnt);
  }
  head1<<<(N_GRAPHS * 128 + ZB - 1) / ZB, ZB, 0, stream>>>(pooled, cnt, l1W, l1b, g1);
  head2<<<(N_GRAPHS * OUT_CH + ZB - 1) / ZB, ZB, 0, stream>>>(g1, l2W, l2b, out);
}